// MixHopConv_4492535791994
// MI455X (gfx1250) — compile-verified
//
#include <hip/hip_runtime.h>

typedef __attribute__((ext_vector_type(2))) float v2f;
typedef __attribute__((ext_vector_type(8))) float v8f;

#define D 128  // feature dim (input and per-hop output)

// ---- degree accumulate: deg[dst[e]] += 1 ----
__global__ __launch_bounds__(256) void k_degree(const int* __restrict__ dst,
                                                float* __restrict__ deg, int E) {
  int e = blockIdx.x * 256 + threadIdx.x;
  if (e < E) {
    __hip_atomic_fetch_add(&deg[dst[e]], 1.0f, __ATOMIC_RELAXED,
                           __HIP_MEMORY_SCOPE_AGENT);
  }
}

// ---- norm = rsqrt(clip(deg,1)) in place ----
__global__ __launch_bounds__(256) void k_norm(float* __restrict__ deg, int N) {
  int i = blockIdx.x * 256 + threadIdx.x;
  if (i < N) {
    float d = deg[i];
    deg[i] = rsqrtf(d < 1.0f ? 1.0f : d);
  }
}

// ---- edge scatter: hnext[dst] += h[src] * norm[src]; one wave per edge ----
__global__ __launch_bounds__(256) void k_scatter(const float* __restrict__ h,
                                                 const float* __restrict__ norm,
                                                 const int* __restrict__ src,
                                                 const int* __restrict__ dst,
                                                 float* __restrict__ hnext, int E) {
  const int lane = threadIdx.x & 31;
  const int e = blockIdx.x * 8 + (threadIdx.x >> 5);
  if (e >= E) return;                       // wave-uniform
  const int s = src[e];
  const int d = dst[e];
  const float ns = norm[s];
  const float4 v = *(const float4*)(h + (size_t)s * D + lane * 4);
  float* o = hnext + (size_t)d * D + lane * 4;
  __hip_atomic_fetch_add(o + 0, v.x * ns, __ATOMIC_RELAXED, __HIP_MEMORY_SCOPE_AGENT);
  __hip_atomic_fetch_add(o + 1, v.y * ns, __ATOMIC_RELAXED, __HIP_MEMORY_SCOPE_AGENT);
  __hip_atomic_fetch_add(o + 2, v.z * ns, __ATOMIC_RELAXED, __HIP_MEMORY_SCOPE_AGENT);
  __hip_atomic_fetch_add(o + 3, v.w * ns, __ATOMIC_RELAXED, __HIP_MEMORY_SCOPE_AGENT);
}

// ---- row scale: h[row][:] *= norm[row]; one wave per row ----
__global__ __launch_bounds__(256) void k_rowscale(float* __restrict__ h,
                                                  const float* __restrict__ norm,
                                                  int N) {
  const int lane = threadIdx.x & 31;
  const int row = blockIdx.x * 8 + (threadIdx.x >> 5);
  if (row >= N) return;                     // wave-uniform
  const float nr = norm[row];
  float4* p = (float4*)(h + (size_t)row * D + lane * 4);
  float4 v = *p;
  v.x *= nr; v.y *= nr; v.z *= nr; v.w *= nr;
  *p = v;
}

// ---- fp32 WMMA GEMM: out[:, colOff:colOff+128] = H (N x 128) @ W (128 x 128)
// One wave computes a 16x128 output strip via 8 accumulator tiles.
// A-frag layout (V_WMMA_F32_16X16X4_F32, ISA 7.12.2): lanes 0-15 hold K=k0,k0+1;
// lanes 16-31 hold K=k0+2,k0+3 (row M = lane&15). B-frag: row-of-B striped
// across lanes per VGPR with the same K half-split.
__global__ __launch_bounds__(256) void k_gemm(const float* __restrict__ H,
                                              const float* __restrict__ W,
                                              float* __restrict__ out,
                                              int N, int ldo, int colOff) {
  const int lane = threadIdx.x & 31;
  const int wave = threadIdx.x >> 5;
  const int rowBase = (blockIdx.x * 8 + wave) * 16;
  if (rowBase >= N) return;                 // wave-uniform: EXEC all-1s for WMMA
  const int half = lane >> 4;               // 0: lanes 0-15, 1: lanes 16-31
  const int l15 = lane & 15;

  v8f acc[8];
  const v8f z = {};
#pragma unroll
  for (int n = 0; n < 8; ++n) acc[n] = z;

  const float* hrow = H + (size_t)(rowBase + l15) * D;
  for (int k0 = 0; k0 < D; k0 += 4) {
    const int kk = k0 + half * 2;
    v2f a;
    a.x = hrow[kk];
    a.y = hrow[kk + 1];
    const float* w0 = W + (size_t)kk * D + l15;
#pragma unroll
    for (int n = 0; n < 8; ++n) {
      v2f b;
      b.x = w0[n * 16];
      b.y = w0[n * 16 + D];
      // 8 args: (neg_a, A, neg_b, B, c_mod, C, reuse_a, reuse_b)
      acc[n] = __builtin_amdgcn_wmma_f32_16x16x4_f32(false, a, false, b,
                                                     (short)0, acc[n],
                                                     false, false);
    }
  }

  // C/D layout: VGPR v -> row rowBase+v (lanes 0-15) / rowBase+v+8 (lanes 16-31),
  // column = lane&15 within each 16-wide tile n.
#pragma unroll
  for (int v = 0; v < 8; ++v) {
    float* orow = out + (size_t)(rowBase + v + half * 8) * ldo + colOff + l15;
#pragma unroll
    for (int n = 0; n < 8; ++n) {
      orow[n * 16] = acc[n][v];
    }
  }
}

extern "C" void kernel_launch(void* const* d_in, const int* in_sizes, int n_in,
                              void* d_out, int out_size, void* d_ws, size_t ws_size,
                              hipStream_t stream) {
  const float* feats = (const float*)d_in[0];
  const float* W0    = (const float*)d_in[1];
  const float* W1    = (const float*)d_in[2];
  const float* W2    = (const float*)d_in[3];
  const int*   src   = (const int*)d_in[4];
  const int*   dst   = (const int*)d_in[5];
  float*       out   = (float*)d_out;

  const int N   = in_sizes[0] / D;   // 100000
  const int E   = in_sizes[4];       // 1600000
  const int ldo = 3 * D;             // 384

  // workspace: [norm: N floats (padded)] [hA: N*128] [hB: N*128]  (~102.8 MB)
  float* norm = (float*)d_ws;
  size_t normElems = ((size_t)N + 255) & ~(size_t)255;
  float* hA = norm + normElems;
  float* hB = hA + (size_t)N * D;

  const int tilesN = (N + 15) / 16;            // 6250 (exact, N % 16 == 0)
  dim3 b256(256);
  dim3 gGemm((tilesN + 7) / 8);                // 8 waves/block, 1 tile-strip/wave
  dim3 gEdgeWave((E + 7) / 8);                 // 1 wave per edge
  dim3 gRowWave((N + 7) / 8);                  // 1 wave per row

  // degree -> norm
  hipMemsetAsync(norm, 0, (size_t)N * sizeof(float), stream);
  k_degree<<<(E + 255) / 256, b256, 0, stream>>>(dst, norm, E);
  k_norm<<<(N + 255) / 256, b256, 0, stream>>>(norm, N);

  // hop 0: out[:, 0:128] = feats @ W0
  k_gemm<<<gGemm, b256, 0, stream>>>(feats, W0, out, N, ldo, 0);

  // hop 1: hA = Dn * A * Dn * feats ; out[:, 128:256] = hA @ W1
  hipMemsetAsync(hA, 0, (size_t)N * D * sizeof(float), stream);
  k_scatter<<<gEdgeWave, b256, 0, stream>>>(feats, norm, src, dst, hA, E);
  k_rowscale<<<gRowWave, b256, 0, stream>>>(hA, norm, N);
  k_gemm<<<gGemm, b256, 0, stream>>>(hA, W1, out, N, ldo, D);

  // hop 2: hB = Dn * A * Dn * hA ; out[:, 256:384] = hB @ W2
  hipMemsetAsync(hB, 0, (size_t)N * D * sizeof(float), stream);
  k_scatter<<<gEdgeWave, b256, 0, stream>>>(hA, norm, src, dst, hB, E);
  k_rowscale<<<gRowWave, b256, 0, stream>>>(hB, norm, N);
  k_gemm<<<gGemm, b256, 0, stream>>>(hB, W2, out, N, ldo, 2 * D);
}